// Learnable_33440615366951
// MI455X (gfx1250) — compile-verified
//
#include <hip/hip_runtime.h>

#define BLOCK 256
#define WAVES_PER_BLOCK 8
#define SLOTS 32            // padded bins per wave (21 used)
#define GRID_BLOCKS 2048

typedef float v2f __attribute__((ext_vector_type(2)));
typedef float v4f __attribute__((ext_vector_type(4)));
typedef float v8f __attribute__((ext_vector_type(8)));

__global__ void zero_bins_kernel(float* __restrict__ g) {
    if (threadIdx.x < SLOTS) g[threadIdx.x] = 0.0f;
}

__global__ __launch_bounds__(BLOCK) void ece_hist_kernel(
    const float* __restrict__ conf, const float* __restrict__ target,
    const float* __restrict__ pa, const float* __restrict__ pb,
    const int* __restrict__ pnb, int n, float* __restrict__ gbins) {
    __shared__ float hist[WAVES_PER_BLOCK * SLOTS];
    const int tid  = threadIdx.x;
    const int wave = tid >> 5;
    const int lane = tid & 31;
    hist[tid] = 0.0f;                       // 256 threads == 8*32 slots
    __syncthreads();

    const float a  = pa[0];
    const float b  = pb[0];
    const int   nb = pnb[0];                // n_bins (20)
    const int maxbin = (nb < SLOTS - 1) ? nb : (SLOTS - 1);
    const float scale = (float)nb;          // 1/step
    float* whist = &hist[wave * SLOTS];

    const long long nvec = (long long)(n >> 2);
    const v4f* c4 = (const v4f*)conf;
    const v4f* t4 = (const v4f*)target;
    long long gid    = (long long)blockIdx.x * BLOCK + tid;
    long long stride = (long long)gridDim.x * BLOCK;

    for (long long i = gid; i < nvec; i += stride) {
        v4f cv = c4[i];
        v4f tv = t4[i];
#pragma unroll
        for (int j = 0; j < 4; ++j) {
            float cf = cv[j];
            float tg = tv[j];
            // logits = -log(1/conf - 1); z = a*logits + b; c = 1/(1+exp(-z))
            float r  = __builtin_amdgcn_rcpf(cf);          // 1/conf
            float l  = __logf(r - 1.0f);                   // -logits
            float e  = __expf(a * l - b);                  // exp(-z)
            float cc = __builtin_amdgcn_rcpf(1.0f + e);    // sigmoid
            int bin  = (int)(cc * scale);                  // cc >= 0: trunc == floor
            bin = bin < 0 ? 0 : (bin > maxbin ? maxbin : bin);
            atomicAdd(&whist[bin], tg - cc);               // ds_add_f32 (per-wave private)
        }
    }
    // scalar tail (n not multiple of 4)
    long long tail = (long long)n - nvec * 4;
    if (gid < tail) {
        float cf = conf[nvec * 4 + gid];
        float tg = target[nvec * 4 + gid];
        float r  = __builtin_amdgcn_rcpf(cf);
        float l  = __logf(r - 1.0f);
        float e  = __expf(a * l - b);
        float cc = __builtin_amdgcn_rcpf(1.0f + e);
        int bin  = (int)(cc * scale);
        bin = bin < 0 ? 0 : (bin > maxbin ? maxbin : bin);
        atomicAdd(&whist[bin], tg - cc);
    }
    __syncthreads();

    // Block reduction via V_WMMA_F32_16X16X4_F32, A = ones:
    //   D[m][n] = sum_k B[k][n]   (layout-agnostic because A is all-ones:
    //   column n of B is exactly the 4 values in lanes {n, n+16} of the two
    //   B VGPRs). Two chained WMMAs sum all 8 wave-histograms for 16 bins;
    //   column group g covers bins g*16 .. g*16+15.
    const int half = lane >> 4;
    const int col  = lane & 15;
    const v2f av = {1.0f, 1.0f};
#pragma unroll
    for (int g = 0; g < 2; ++g) {
        int binc = g * 16 + col;
        v2f b01 = { hist[(2 * half + 0) * SLOTS + binc],
                    hist[(2 * half + 1) * SLOTS + binc] };
        v2f b23 = { hist[(2 * half + 4) * SLOTS + binc],
                    hist[(2 * half + 5) * SLOTS + binc] };
        v8f acc = {};
        acc = __builtin_amdgcn_wmma_f32_16x16x4_f32(
                  false, av, false, b01, (short)0, acc, false, false);
        acc = __builtin_amdgcn_wmma_f32_16x16x4_f32(
                  false, av, false, b23, (short)0, acc, false, false);
        // every lane's acc[0] holds the column-sum for its col; one wave commits
        if (wave == 0 && lane < 16 && binc <= maxbin) {
            atomicAdd(&gbins[binc], acc[0]);
        }
    }
}

__global__ void ece_final_kernel(const float* __restrict__ gbins,
                                 const int* __restrict__ pnb,
                                 int n, float* __restrict__ out) {
    if (threadIdx.x == 0) {
        int cnt = pnb[0] + 1;
        if (cnt > SLOTS) cnt = SLOTS;
        float s = 0.0f;
        for (int i = 0; i < cnt; ++i) s += fabsf(gbins[i]);
        out[0] = s / (float)n;
    }
}

extern "C" void kernel_launch(void* const* d_in, const int* in_sizes, int n_in,
                              void* d_out, int out_size, void* d_ws, size_t ws_size,
                              hipStream_t stream) {
    const float* conf   = (const float*)d_in[0];
    const float* target = (const float*)d_in[1];
    const float* pa     = (const float*)d_in[2];
    const float* pb     = (const float*)d_in[3];
    const int*   pnb    = (const int*)d_in[4];
    const int n = in_sizes[0];

    float* gbins = (float*)d_ws;      // 32 floats of scratch
    float* out   = (float*)d_out;

    zero_bins_kernel<<<1, 32, 0, stream>>>(gbins);
    ece_hist_kernel<<<GRID_BLOCKS, BLOCK, 0, stream>>>(conf, target, pa, pb, pnb, n, gbins);
    ece_final_kernel<<<1, 32, 0, stream>>>(gbins, pnb, n, out);
}